// ClashLoss_89292370084241
// MI455X (gfx1250) — compile-verified
//
#include <hip/hip_runtime.h>

typedef __attribute__((ext_vector_type(2))) float v2f;
typedef __attribute__((ext_vector_type(8))) float v8f;

#define NATOMS 4096
#define TILES  (NATOMS / 16)   // 256 tiles per batch
#define EPS2   1e-16f          // EPS = 1e-8 squared (compare in d^2 domain)

__device__ __forceinline__ unsigned wave_mask32(bool pred) {
#if __has_builtin(__builtin_amdgcn_ballot_w32)
    return __builtin_amdgcn_ballot_w32(pred);   // folds into the v_cmp itself
#else
    return (unsigned)__ballot(pred);
#endif
}

__global__ void ClashLoss_zero_kernel(unsigned int* cnt) {
    cnt[0] = 0u;
}

// One wave per (batch, 16-row i-tile); 8 waves per block.
// LDS holds the batch pre-packed in WMMA B-operand layout:
//   bp[2a+0] = (-2x, -2y)   -> B K=0,1 rows (VGPR0/VGPR1, lanes 0-15)
//   bp[2a+1] = (-2z,  1 )   -> B K=2,3 rows (VGPR0/VGPR1, lanes 16-31)
//   sr2[a]   = (|c|^2, r)
// A row = [x, y, z, |ci|^2], B col = [-2xj, -2yj, -2zj, 1]
//   => WMMA acc = |ci|^2 - 2 ci.cj ; clash test done directly on acc:
//   acc > EPS2 - sqj   &&   acc < (ri+rj)^2 - sqj.
__global__ __launch_bounds__(256) void ClashLoss_main_kernel(
    const float* __restrict__ coords,   // [B, N, 3]
    const int*   __restrict__ types,    // [B, N]
    const float* __restrict__ vdw,      // [T]
    unsigned int* __restrict__ count)
{
    extern __shared__ v2f smem[];
    v2f* bp  = smem;                    // [2*NATOMS]
    v2f* sr2 = smem + 2 * NATOMS;       // [NATOMS]

    const int blocks_per_batch = TILES / 8;            // 32
    const int b     = blockIdx.x / blocks_per_batch;
    const int tile0 = (blockIdx.x % blocks_per_batch) * 8;

    const float* cb = coords + (size_t)b * NATOMS * 3;
    const int*   tb = types  + (size_t)b * NATOMS;

    // Cooperative stage of the whole batch into LDS.
    for (int a = threadIdx.x; a < NATOMS; a += 256) {
        const float x = cb[a * 3 + 0];
        const float y = cb[a * 3 + 1];
        const float z = cb[a * 3 + 2];
        v2f p0; p0.x = -2.0f * x; p0.y = -2.0f * y;
        v2f p1; p1.x = -2.0f * z; p1.y = 1.0f;
        bp[2 * a + 0] = p0;
        bp[2 * a + 1] = p1;
        v2f s; s.x = x * x + y * y + z * z; s.y = vdw[tb[a]];
        sr2[a] = s;
    }
    __syncthreads();

    const int wave   = threadIdx.x >> 5;
    const int lane   = threadIdx.x & 31;
    const int halfid = lane >> 4;       // 0: lanes 0-15 (K=0,1), 1: lanes 16-31 (K=2,3)
    const int m      = lane & 15;

    const int ai = (tile0 + wave) * 16 + m;
    const v2f bi0 = bp[2 * ai + 0];     // (-2x, -2y)
    const v2f bi1 = bp[2 * ai + 1];     // (-2z,  1 )
    const v2f si  = sr2[ai];            // (sq, r)

    // A-matrix 16x4 f32: lanes 0-15 carry K=0,1 = (x, y);
    // lanes 16-31 carry K=2,3 = (z, sq_i).
    v2f A;
    A.x = halfid ? (-0.5f * bi1.x) : (-0.5f * bi0.x);
    A.y = halfid ? si.x            : (-0.5f * bi0.y);

    // Per-row radius for the D layout (row M = v + 8*halfid).
    const float ri = si.y;
    float riv[8];
#pragma unroll
    for (int v = 0; v < 8; ++v)
        riv[v] = __shfl(ri, v + 8 * halfid, 32);

    unsigned int cnt = 0;               // wave-uniform via ballot counting

    for (int jt = 0; jt < TILES; ++jt) {
        const int aj = jt * 16 + m;
        const v2f Bm = bp[2 * aj + halfid];   // conflict-free ds_load_b64
        const v2f sj = sr2[aj];               // (sq_j, r_j), broadcast across halves

        v8f acc = {};
        acc = __builtin_amdgcn_wmma_f32_16x16x4_f32(
            /*neg_a=*/false, A, /*neg_b=*/false, Bm,
            /*c_mod=*/(short)0, acc, /*reuse_a=*/false, /*reuse_b=*/false);

        const float lo = EPS2 - sj.x;          // per-lane, once per j-tile

#pragma unroll
        for (int v = 0; v < 8; ++v) {
            const float md = riv[v] + sj.y;
            const float hi = __builtin_fmaf(md, md, -sj.x);  // md^2 - sqj
            // (d2 > EPS2 && d2 < md^2) with d2 = acc + sqj; clamp subsumed by lo.
            const unsigned m1 = wave_mask32(acc[v] > lo);
            const unsigned m2 = wave_mask32(acc[v] < hi);
            cnt += (unsigned)__builtin_popcount(m1 & m2);    // s_and + s_bcnt1
        }
    }

    if (lane == 0)
        atomicAdd(count, cnt);
}

__global__ void ClashLoss_final_kernel(const unsigned int* cnt, float* out, float scale) {
    out[0] = (float)cnt[0] * scale;
}

extern "C" void kernel_launch(void* const* d_in, const int* in_sizes, int n_in,
                              void* d_out, int out_size, void* d_ws, size_t ws_size,
                              hipStream_t stream) {
    const float* coords = (const float*)d_in[0];   // [B, N, 3] f32
    const int*   types  = (const int*)d_in[1];     // [B, N] int
    const float* vdw    = (const float*)d_in[2];   // [T] f32

    const int B = in_sizes[0] / (NATOMS * 3);      // = 8
    unsigned int* cnt = (unsigned int*)d_ws;
    float* out = (float*)d_out;

    ClashLoss_zero_kernel<<<1, 1, 0, stream>>>(cnt);

    const int blocks = B * (TILES / 8);            // 256 blocks x 256 threads
    const size_t lds_bytes = (size_t)(3 * NATOMS) * sizeof(v2f);  // 96 KB
    ClashLoss_main_kernel<<<blocks, 256, lds_bytes, stream>>>(coords, types, vdw, cnt);

    const float scale = 1.0f / (2.0f * (float)NATOMS * (float)B);
    ClashLoss_final_kernel<<<1, 1, 0, stream>>>(cnt, out, scale);
}